// VanillaClassifierStableV2_34471407518048
// MI455X (gfx1250) — compile-verified
//
#include <hip/hip_runtime.h>
#include <hip/hip_bf16.h>
#include <math.h>

typedef float v2f __attribute__((ext_vector_type(2)));
typedef float v8f __attribute__((ext_vector_type(8)));

// ---------------------------------------------------------------------------
// Online-softmax update: keep running max m and running sum s of exp(x - m).
// Branchless: one of the two expf args is 0 -> expf(0)=1.
// ---------------------------------------------------------------------------
__device__ __forceinline__ void ol_update(float& m, float& s, float x) {
    float mn = fmaxf(m, x);
    s = s * expf(m - mn) + expf(x - mn);   // expf(-inf) == 0 handles first elem
    m = mn;
}

__device__ __forceinline__ void ol_combine(float& m1, float& s1, float m2, float s2) {
    float mn = fmaxf(m1, m2);
    s1 = s1 * expf(m1 - mn) + s2 * expf(m2 - mn);
    m1 = mn;
}

// ---------------------------------------------------------------------------
// Kernel 1: one row (C logits) per workgroup. Single pass over memory
// (online softmax), B128 vector loads after an alignment prologue, prefetch
// ahead, LDS tree reduction, per-row loss -> workspace.
// Roofline: 4096 x 50257 x 4B = 823 MB read once -> ~35 us at 23.3 TB/s.
// ---------------------------------------------------------------------------
#define BLK 256

__global__ __launch_bounds__(BLK) void ce_row_kernel(
    const float* __restrict__ pred,
    const int*   __restrict__ target,
    float*       __restrict__ rowloss,
    int C)
{
    const int b   = blockIdx.x;
    const int tid = threadIdx.x;
    const float* __restrict__ row = pred + (size_t)b * (size_t)C;

    float m = -INFINITY;
    float s = 0.0f;

    // Alignment prologue: C is odd, so rows are only 4B aligned. Peel scalars
    // until 16B-aligned for float4 (B128) streaming.
    const uintptr_t addr = (uintptr_t)row;
    const int lead = (int)(((16u - (addr & 15u)) & 15u) >> 2);   // 0..3 floats
    if (tid < lead && tid < C) ol_update(m, s, row[tid]);

    const int vbase = (lead < C) ? lead : C;
    const int nvec  = (C - vbase) >> 2;                 // # of float4 chunks
    const float4* __restrict__ row4 = (const float4*)(row + vbase);

    for (int i = tid; i < nvec; i += BLK) {
        if (i + 2 * BLK < nvec)
            __builtin_prefetch(&row4[i + 2 * BLK], 0, 0);  // global_prefetch_b8
        float4 v = row4[i];
        ol_update(m, s, v.x);
        ol_update(m, s, v.y);
        ol_update(m, s, v.z);
        ol_update(m, s, v.w);
    }

    // Scalar tail
    const int tail_base = vbase + (nvec << 2);
    const int rem = C - tail_base;
    if (tid < rem) ol_update(m, s, row[tail_base + tid]);

    // Block-wide (m, s) reduction in LDS
    __shared__ float sm[BLK];
    __shared__ float ss[BLK];
    sm[tid] = m; ss[tid] = s;
    __syncthreads();
    for (int off = BLK >> 1; off > 0; off >>= 1) {
        if (tid < off) {
            float m1 = sm[tid], s1 = ss[tid];
            ol_combine(m1, s1, sm[tid + off], ss[tid + off]);
            sm[tid] = m1; ss[tid] = s1;
        }
        __syncthreads();
    }

    if (tid == 0) {
        float mt = sm[0];
        float st = ss[0];
        int   t  = target[b];
        float xt = row[t];
        // -log_softmax(x)[t] = m + log(sum exp(x-m)) - x_t
        rowloss[b] = mt + logf(st) - xt;
    }
}

// ---------------------------------------------------------------------------
// Kernel 2: deterministic sum of n per-row losses via the matrix unit.
// One wave32. D = A x B + C with B = all-ones:
//   D[m][n] = sum_k A[m][k] + C[m][n]  -> 64 values folded per WMMA issue.
// A (16x4 f32) layout: lanes 0-15 hold M=lane, K=0/1 in VGPR0/1;
//                      lanes 16-31 hold M=lane-16, K=2/3.
// Each lane's A fragment is 2 contiguous floats -> one unguarded B64 load
// (full 64-element chunks only; remainder handled by a butterfly tail), so
// no EXEC masking anywhere near the WMMA ops.
// C/D (16x16 f32) layout: VGPR j, lanes 0-15 -> M=j, N=lane;
//                                  lanes 16-31 -> M=8+j, N=lane-16.
// So lane 0 sums M=0..7 at N=0, lane 16 sums M=8..15 at N=0.
// ---------------------------------------------------------------------------
__global__ __launch_bounds__(32) void reduce_wmma_kernel(
    const float* __restrict__ vals,
    float*       __restrict__ out,
    int n, float scale)
{
    const int lane = threadIdx.x;          // 0..31, single wave
    const int rowm = lane & 15;            // M index
    const int koff = (lane >> 4) << 1;     // K base: 0 for lanes 0-15, 2 else
    const int aoff = rowm * 4 + koff;      // lane's element offset in a chunk

    v2f bones; bones[0] = 1.0f; bones[1] = 1.0f;
    v8f acc = {};

    const int nfull = n & ~63;             // full 64-element chunks
    for (int base = 0; base < nfull; base += 64) {
        // 8B-aligned contiguous pair -> global_load_b64, no guards
        v2f a = *(const v2f*)(vals + base + aoff);
        // 8 args: (neg_a, A, neg_b, B, c_mod, C, reuse_a, reuse_b)
        acc = __builtin_amdgcn_wmma_f32_16x16x4_f32(
            false, a, false, bones, (short)0, acc, false, false);
    }

    // Per-lane: sum its 8 accumulator rows (all at this lane's N column).
    float part = acc[0] + acc[1] + acc[2] + acc[3]
               + acc[4] + acc[5] + acc[6] + acc[7];
    // lane 0 has M=0..7 @ N=0, lane 16 has M=8..15 @ N=0: combine.
    float other = __shfl(part, lane ^ 16, 32);

    // Remainder (n & 63): scalar per-lane + fixed-order butterfly (determin.)
    float t = 0.0f;
    for (int i = nfull + lane; i < n; i += 32) t += vals[i];
    #pragma unroll
    for (int off = 16; off > 0; off >>= 1) t += __shfl_xor(t, off, 32);

    if (lane == 0) out[0] = (part + other + t) * scale;
}

// ---------------------------------------------------------------------------
// Launch: d_in[0] = pred [B*C] f32, d_in[1] = target [B] int,
// d_out[0] = scalar f32 loss, d_ws = B floats of scratch.
// ---------------------------------------------------------------------------
extern "C" void kernel_launch(void* const* d_in, const int* in_sizes, int n_in,
                              void* d_out, int out_size, void* d_ws, size_t ws_size,
                              hipStream_t stream) {
    const float* pred   = (const float*)d_in[0];
    const int*   target = (const int*)d_in[1];
    float*       out    = (float*)d_out;
    float*       rowls  = (float*)d_ws;

    const int B = in_sizes[1];              // 4096
    const int C = in_sizes[0] / B;          // 50257

    ce_row_kernel<<<B, BLK, 0, stream>>>(pred, target, rowls, C);
    reduce_wmma_kernel<<<1, 32, 0, stream>>>(rowls, out, B, 1.0f / (float)B);
}